// GCC_GraphControl_KHop_76055280878126
// MI455X (gfx1250) — compile-verified
//
#include <hip/hip_runtime.h>
#include <hip/hip_bf16.h>

// ---------------------------------------------------------------------------
// GCN forward on gfx1250 (MI455X).
// Dense GEMMs: V_WMMA_F32_16X16X4_F32, K/OC/FLAGS compile-time: K loop fully
// unrolls, bound checks and epilogue flag tests constant-fold, loads are
// clamp+select (branch-free), stores take a wave-uniform full-tile fast path.
// Sparse propagation: float4-vectorized edge-parallel atomic scatter; whole
// working set (~26MB/matrix) is L2-resident on the 192MB L2.
// ---------------------------------------------------------------------------

typedef float v2f __attribute__((ext_vector_type(2)));
typedef float v8f __attribute__((ext_vector_type(8)));

#define HDIM 128
#define PDIM 32

// ---------------- elementwise / graph kernels ----------------

__global__ void k_fill(float* p, float v, long n) {
  long i = (long)blockIdx.x * blockDim.x + threadIdx.x;
  if (i < n) p[i] = v;
}

__global__ void k_deg(const int* __restrict__ dst, float* deg, int E) {
  int e = blockIdx.x * blockDim.x + threadIdx.x;
  if (e < E) atomicAdd(&deg[dst[e]], 1.0f);
}

__global__ void k_dinv(float* d, int n) {
  int i = blockIdx.x * blockDim.x + threadIdx.x;
  if (i < n) d[i] = rsqrtf(fmaxf(d[i], 1.0f));
}

__global__ void k_edge_w(const int* __restrict__ src, const int* __restrict__ dst,
                         const float* __restrict__ dinv, float* w, int E) {
  int e = blockIdx.x * blockDim.x + threadIdx.x;
  if (e < E) w[e] = dinv[src[e]] * dinv[dst[e]];
}

__global__ void k_flag(const int* __restrict__ root, float* flag, int g) {
  int i = blockIdx.x * blockDim.x + threadIdx.x;
  if (i < g) flag[root[i]] = 1.0f;
}

// h0 = concat(x_pe, flag, zeros) into both h_f and h_c (stride 128)
__global__ void k_init_h(const float* __restrict__ xpe, const float* __restrict__ flag,
                         float* h_f, float* h_c, long total) {
  long idx = (long)blockIdx.x * blockDim.x + threadIdx.x;
  if (idx >= total) return;
  long n = idx >> 7; int f = (int)(idx & 127);
  float v = (f < PDIM) ? xpe[n * PDIM + f] : ((f == PDIM) ? flag[n] : 0.0f);
  h_f[idx] = v; h_c[idx] = v;
}

__global__ void k_copy(float* __restrict__ d, const float* __restrict__ s, long n) {
  long i = (long)blockIdx.x * blockDim.x + threadIdx.x;
  if (i < n) d[i] = s[i];
}

// self-loop contribution: out[n,f] = h[n,f] * dinv[n]^2
__global__ void k_prop_init(const float* __restrict__ h, const float* __restrict__ dinv,
                            float* __restrict__ out, long total) {
  long idx = (long)blockIdx.x * blockDim.x + threadIdx.x;
  if (idx >= total) return;
  long n = idx >> 7;
  float di = dinv[n];
  out[idx] = h[idx] * di * di;
}

// edge scatter: out[dst, 4q:4q+4] += h[src, 4q:4q+4] * w[e]
// one thread per (edge, 4 features): float4 gather (global_load_b128) + 4 atomics
__global__ void k_prop_edges(const float* __restrict__ h, const int* __restrict__ src,
                             const int* __restrict__ dst, const float* __restrict__ w,
                             float* __restrict__ out, int E) {
  long idx = (long)blockIdx.x * blockDim.x + threadIdx.x;
  long e = idx >> 5; int q = (int)(idx & 31);
  if (e >= E) return;
  const int s = src[e], d = dst[e];
  const float we = w[e];
  const float4 hv = *(const float4*)(h + (long)s * HDIM + q * 4);
  float* o = out + (long)d * HDIM + q * 4;
  atomicAdd(o + 0, hv.x * we);
  atomicAdd(o + 1, hv.y * we);
  atomicAdd(o + 2, hv.z * we);
  atomicAdd(o + 3, hv.w * we);
}

__global__ void k_relu_bias(const float* __restrict__ p, const float* __restrict__ b,
                            float* __restrict__ out, long total) {
  long idx = (long)blockIdx.x * blockDim.x + threadIdx.x;
  if (idx >= total) return;
  int f = (int)(idx & 127);
  out[idx] = fmaxf(p[idx] + b[f], 0.0f);
}

// pe[n, 0:32] = thresholded pe_h[n, 0:32]
__global__ void k_thresh(const float* __restrict__ pe_h, float* __restrict__ pe, long total) {
  long idx = (long)blockIdx.x * blockDim.x + threadIdx.x;
  if (idx >= total) return;
  long n = idx >> 5; int j = (int)(idx & 31);
  float v = pe_h[n * HDIM + j];
  pe[idx] = (fabsf(v) > 0.1f) ? v : 0.0f;
}

// ca[n,k] += h_c[n,k] for k < Kin (stride 128)
__global__ void k_add(float* __restrict__ ca, const float* __restrict__ hc, int Kin, long total) {
  long idx = (long)blockIdx.x * blockDim.x + threadIdx.x;
  if (idx >= total) return;
  int f = (int)(idx & 127);
  if (f < Kin) ca[idx] += hc[idx];
}

__global__ void k_accum(float* __restrict__ acc, const float* __restrict__ h, long n) {
  long i = (long)blockIdx.x * blockDim.x + threadIdx.x;
  if (i < n) acc[i] += h[i];
}

__global__ void k_counts(const int* __restrict__ batch, float* counts, int N) {
  int i = blockIdx.x * blockDim.x + threadIdx.x;
  if (i < N) atomicAdd(&counts[batch[i]], 1.0f);
}

__global__ void k_pool(const float* __restrict__ hs, const int* __restrict__ batch,
                       float* __restrict__ pooled, long total) {
  long idx = (long)blockIdx.x * blockDim.x + threadIdx.x;
  if (idx >= total) return;
  long n = idx >> 7; int f = (int)(idx & 127);
  atomicAdd(&pooled[(long)batch[n] * HDIM + f], hs[idx]);
}

// per-graph: mean, L2-normalize, classify.  1 block(128) per graph.
__global__ void k_final(const float* __restrict__ pooled, const float* __restrict__ counts,
                        const float* __restrict__ W, const float* __restrict__ b,
                        float* __restrict__ out, int C) {
  __shared__ float sv[HDIM];
  __shared__ float red[HDIM];
  int g = blockIdx.x, f = threadIdx.x;
  float inv = 1.0f / fmaxf(counts[g], 1.0f);
  float v = pooled[(long)g * HDIM + f] * inv;
  sv[f] = v; red[f] = v * v;
  __syncthreads();
  for (int s = 64; s > 0; s >>= 1) {
    if (f < s) red[f] += red[f + s];
    __syncthreads();
  }
  float scale = 1.0f / fmaxf(sqrtf(red[0]), 1e-5f);
  if (f < C) {
    float acc = 0.0f;
    for (int k = 0; k < HDIM; ++k) acc += sv[k] * scale * W[k * C + f];
    out[(long)g * C + f] = acc + b[f];
  }
}

// ---------------- fp32 WMMA GEMM (compile-time K, OC, FLAGS) ----------------
// C[nrows x OC] = op( [addend +] A[nrows x K] @ W[K x OC] [+ bias] )
// One wave per 16-row strip; 16x16 output tiles; K-step 4, fully unrolled.
// Loads are unconditional (indices clamped, results masked with selects).
// Stores: wave-uniform full-tile fast path (nrows is a multiple of 16 in
// practice) -> 8 straight global_store_b32 per tile, no exec-mask branches.
// REQUIREMENT: KP <= lda at every call site (holds: 36<=128, 32<=32, 128<=128).
// FLAGS: 1 = +bias, 2 = +addend, 4 = relu
template <int K, int OC, int FLAGS>
__global__ void k_wmma_gemm(const float* __restrict__ A, int lda,
                            const float* __restrict__ W,
                            const float* __restrict__ bias,
                            const float* __restrict__ addend,
                            float* __restrict__ C, int ldc,
                            int nrows) {
  constexpr int KP  = (K + 3) & ~3;   // K padded to multiple of 4
  constexpr int NCT = (OC + 15) / 16; // 16-col output tiles
  const int lane = threadIdx.x & 31;
  const int wid  = threadIdx.x >> 5;
  const int strip = blockIdx.x * (blockDim.x >> 5) + wid;
  if (strip * 16 >= nrows) return;            // wave-uniform exit (EXEC stays full)
  const int half = lane >> 4;                 // 0: lanes 0-15, 1: lanes 16-31
  const int m    = lane & 15;
  const int arow = strip * 16 + m;
  const bool arow_ok = (arow < nrows);
  const int asafe = arow_ok ? arow : 0;
  const bool full = (strip * 16 + 16 <= nrows);  // wave-uniform

  // Preload all A fragments for this strip (reused across all column tiles).
  // A 16x4 fp32 fragment: lanes 0-15 hold K={k0,k0+1}, lanes 16-31 K={k0+2,k0+3}.
  v2f afr[KP / 4];
#pragma unroll
  for (int s = 0; s < KP / 4; ++s) {
    const int ka = s * 4 + half * 2;
    float x = A[(long)asafe * lda + ka];       // safe: ka < KP <= lda
    float y = A[(long)asafe * lda + ka + 1];
    if (ka     >= K) x = 0.0f;                 // folds away when K % 4 == 0
    if (ka + 1 >= K) y = 0.0f;
    afr[s].x = arow_ok ? x : 0.0f;
    afr[s].y = arow_ok ? y : 0.0f;
  }

  for (int ct = 0; ct < NCT; ++ct) {
    const int col0 = ct << 4;
    const int bcol = col0 + m;
    const bool bcol_ok = (bcol < OC);          // folds when OC % 16 == 0
    const int csafe = bcol_ok ? bcol : (OC - 1);
    v8f acc = {0.f, 0.f, 0.f, 0.f, 0.f, 0.f, 0.f, 0.f};
#pragma unroll
    for (int s = 0; s < KP / 4; ++s) {
      const int ka  = s * 4 + half * 2;
      const int k0s = (ka     < K) ? ka     : (K - 1);
      const int k1s = (ka + 1 < K) ? ka + 1 : (K - 1);
      v2f bfr;
      bfr.x = W[(long)k0s * OC + csafe];       // unconditional, clamped
      bfr.y = W[(long)k1s * OC + csafe];
      if (ka     >= K) bfr.x = 0.0f;           // folds when K % 4 == 0
      if (ka + 1 >= K) bfr.y = 0.0f;
      if (!bcol_ok) { bfr.x = 0.0f; bfr.y = 0.0f; }
      acc = __builtin_amdgcn_wmma_f32_16x16x4_f32(false, afr[s], false, bfr,
                                                  (short)0, acc, false, false);
    }

    // Epilogue. D layout: VGPR r -> (M=r, N=lane) lanes 0-15 ;
    //                               (M=r+8, N=lane-16) lanes 16-31.
    float bv = 0.0f;
    if (FLAGS & 1) bv = bias[csafe];           // one load per tile (folds if unused)
    float vals[8];
#pragma unroll
    for (int r = 0; r < 8; ++r) {
      const int orow  = strip * 16 + r + half * 8;
      const int osafe = (orow < nrows) ? orow : 0;
      float v = acc[r];
      if (FLAGS & 1) v += bv;                  // no +0.0f in FLAGS==0 variants
      if (FLAGS & 2) v += addend[(long)osafe * ldc + csafe];  // clamped, unconditional
      if (FLAGS & 4) v = fmaxf(v, 0.f);
      vals[r] = v;
    }
    if ((OC % 16 == 0) && full) {
      // fast path: unconditional stores (wave-uniform branch)
#pragma unroll
      for (int r = 0; r < 8; ++r)
        C[(long)(strip * 16 + r + half * 8) * ldc + bcol] = vals[r];
    } else {
#pragma unroll
      for (int r = 0; r < 8; ++r) {
        const int orow = strip * 16 + r + half * 8;
        if (orow < nrows && bcol_ok) C[(long)orow * ldc + bcol] = vals[r];
      }
    }
  }
}

static inline void gemm_launch(int K, int OC, int flags, const float* A, int lda,
                               const float* Wt, const float* bias, const float* add,
                               float* Cc, int ldc, int nrows, unsigned gblocks,
                               hipStream_t stream) {
  if (K == 33 && OC == 128 && flags == 0)
    k_wmma_gemm<33, 128, 0><<<gblocks, 128, 0, stream>>>(A, lda, Wt, bias, add, Cc, ldc, nrows);
  else if (K == 128 && OC == 128 && flags == 0)
    k_wmma_gemm<128, 128, 0><<<gblocks, 128, 0, stream>>>(A, lda, Wt, bias, add, Cc, ldc, nrows);
  else if (K == 32 && OC == 128 && flags == 1)
    k_wmma_gemm<32, 128, 1><<<gblocks, 128, 0, stream>>>(A, lda, Wt, bias, add, Cc, ldc, nrows);
  else if (K == 128 && OC == 33 && flags == 1)
    k_wmma_gemm<128, 33, 1><<<gblocks, 128, 0, stream>>>(A, lda, Wt, bias, add, Cc, ldc, nrows);
  else if (K == 128 && OC == 128 && flags == 1)
    k_wmma_gemm<128, 128, 1><<<gblocks, 128, 0, stream>>>(A, lda, Wt, bias, add, Cc, ldc, nrows);
  else if (K == 128 && OC == 128 && flags == 3)
    k_wmma_gemm<128, 128, 3><<<gblocks, 128, 0, stream>>>(A, lda, Wt, bias, add, Cc, ldc, nrows);
}

// ---------------- host driver ----------------

extern "C" void kernel_launch(void* const* d_in, const int* in_sizes, int n_in,
                              void* d_out, int out_size, void* d_ws, size_t ws_size,
                              hipStream_t stream) {
  (void)n_in; (void)out_size; (void)ws_size;
  const int L = 5, C = 40;
  const float* x_pe   = (const float*)d_in[0];
  const float* x_orig = (const float*)d_in[1];
  const int*   eidx   = (const int*)d_in[2];
  const int*   batch  = (const int*)d_in[3];
  const int*   root   = (const int*)d_in[4];
  const int N = in_sizes[3];
  const int E = in_sizes[2] / 2;
  const int G = in_sizes[4];
  const int* srcE = eidx;
  const int* dstE = eidx + E;

  const float *encW[5], *encB[5], *ctrlW[5], *ctrlB[5], *cpW[5], *cpB[5],
              *caW[5], *caB[5], *zW[5], *zB[5];
  for (int i = 0; i < L; ++i) {
    encW[i]  = (const float*)d_in[5 + i];   encB[i]  = (const float*)d_in[10 + i];
    ctrlW[i] = (const float*)d_in[15 + i];  ctrlB[i] = (const float*)d_in[20 + i];
    cpW[i]   = (const float*)d_in[25 + i];  cpB[i]   = (const float*)d_in[30 + i];
    caW[i]   = (const float*)d_in[35 + i];  caB[i]   = (const float*)d_in[40 + i];
    zW[i]    = (const float*)d_in[45 + i];  zB[i]    = (const float*)d_in[50 + i];
  }
  const float* clfW = (const float*)d_in[55];
  const float* clfB = (const float*)d_in[56];

  // workspace carve-out (fp32)
  float* ws = (float*)d_ws;
  size_t off = 0;
  auto carve = [&](size_t n) { float* p = ws + off; off += n; return p; };
  const long NH = (long)N * HDIM;
  float* dinv   = carve(N);
  float* wE     = carve(E);
  float* flag   = carve(N);
  float* h_f    = carve(NH);
  float* h_c    = carve(NH);
  float* pe_h   = carve(NH);
  float* pe_h2  = carve(NH);
  float* t      = carve(NH);
  float* p      = carve(NH);
  float* ca     = carve(NH);
  float* pe     = carve((size_t)N * PDIM);
  float* hid    = carve(NH);
  float* pooled = carve((size_t)G * HDIM);
  float* counts = carve(G);

  auto cdiv = [](long a, long b) { return (unsigned)((a + b - 1) / b); };
  const dim3 blk(256);

  // GCN edge weights: deg (with self loop) -> dinv -> per-edge weight
  k_fill<<<cdiv(N, 256), blk, 0, stream>>>(dinv, 1.0f, N);
  k_deg<<<cdiv(E, 256), blk, 0, stream>>>(dstE, dinv, E);
  k_dinv<<<cdiv(N, 256), blk, 0, stream>>>(dinv, N);
  k_edge_w<<<cdiv(E, 256), blk, 0, stream>>>(srcE, dstE, dinv, wE, E);

  // h0 = [x_pe | flag | 0...] -> h_f, h_c ; pe_h = x_original ; hid = 0
  k_fill<<<cdiv(N, 256), blk, 0, stream>>>(flag, 0.0f, N);
  k_flag<<<cdiv(G, 256), blk, 0, stream>>>(root, flag, G);
  k_init_h<<<cdiv(NH, 256), blk, 0, stream>>>(x_pe, flag, h_f, h_c, NH);
  k_copy<<<cdiv(NH, 256), blk, 0, stream>>>(pe_h, x_orig, NH);
  k_fill<<<cdiv(NH, 256), blk, 0, stream>>>(hid, 0.0f, NH);

  const int strips = (N + 15) / 16;
  const unsigned gblocks = (unsigned)((strips + 3) / 4);   // 4 waves / block
  auto gemm = [&](int K, int OC, int flags, const float* A, int lda, const float* Wt,
                  const float* bias, const float* add, float* Cc, int ldc) {
    gemm_launch(K, OC, flags, A, lda, Wt, bias, add, Cc, ldc, N, gblocks, stream);
  };
  auto prop = [&](const float* hin, float* hout) {
    k_prop_init<<<cdiv(NH, 256), blk, 0, stream>>>(hin, dinv, hout, NH);
    k_prop_edges<<<cdiv((long)E * 32, 256), blk, 0, stream>>>(hin, srcE, dstE, wE, hout, E);
  };

  for (int i = 0; i < L; ++i) {
    const int Kin  = i ? HDIM : 33;   // h_f / h_c feature width into layer
    const int OCad = i ? HDIM : 33;   // cond_adapt output width

    // h_f = relu(prop(h_f @ enc_W) + enc_b)
    gemm(Kin, HDIM, 0, h_f, HDIM, encW[i], nullptr, nullptr, t, HDIM);
    prop(t, p);
    k_relu_bias<<<cdiv(NH, 256), blk, 0, stream>>>(p, encB[i], h_f, NH);

    // pe = threshold(pe_h[:, :32]); cond = pe@cpW+b; cond_ad = cond@caW+b
    k_thresh<<<cdiv((long)N * PDIM, 256), blk, 0, stream>>>(pe_h, pe, (long)N * PDIM);
    gemm(PDIM, HDIM, 1, pe, PDIM, cpW[i], cpB[i], nullptr, t, HDIM);
    gemm(HDIM, OCad, 1, t, HDIM, caW[i], caB[i], nullptr, ca, HDIM);

    // h_c = relu(prop((h_c + cond_ad) @ ctrl_W) + ctrl_b)
    k_add<<<cdiv(NH, 256), blk, 0, stream>>>(ca, h_c, Kin, NH);
    gemm(Kin, HDIM, 0, ca, HDIM, ctrlW[i], nullptr, nullptr, t, HDIM);
    prop(t, p);
    k_relu_bias<<<cdiv(NH, 256), blk, 0, stream>>>(p, ctrlB[i], h_c, NH);

    // h_f = h_f + h_c @ zero_W + zero_b ; hid += h_f
    gemm(HDIM, HDIM, 3, h_c, HDIM, zW[i], zB[i], h_f, h_f, HDIM);
    k_accum<<<cdiv(NH, 256), blk, 0, stream>>>(hid, h_f, NH);

    // pe_h = prop(pe_h)
    prop(pe_h, pe_h2);
    float* tmp = pe_h; pe_h = pe_h2; pe_h2 = tmp;
  }

  // pooling: pooled[g] = mean over nodes of sum_l h_l ; normalize ; classify
  k_fill<<<cdiv(G, 256), blk, 0, stream>>>(counts, 0.0f, G);
  k_fill<<<cdiv((long)G * HDIM, 256), blk, 0, stream>>>(pooled, 0.0f, (long)G * HDIM);
  k_counts<<<cdiv(N, 256), blk, 0, stream>>>(batch, counts, N);
  k_pool<<<cdiv(NH, 256), blk, 0, stream>>>(hid, batch, pooled, NH);
  k_final<<<G, HDIM, 0, stream>>>(pooled, counts, clfW, clfB, (float*)d_out, C);
}